// LocalTrittention_28166395527527
// MI455X (gfx1250) — compile-verified
//
#include <hip/hip_runtime.h>

typedef __attribute__((ext_vector_type(2))) float v2f;
typedef __attribute__((ext_vector_type(8))) float v8f;

// D = A(16x4) * B(4x16) + C, fp32, wave32.
// A/B fragment: VGPR j, lane L -> K = j + 2*(L>=16), M/N = L&15.
// C/D: VGPR i, lane L -> M = i + 8*(L>=16), N = L&15.
#define WMMA_F32(a, b, c) \
  __builtin_amdgcn_wmma_f32_16x16x4_f32(false, (a), false, (b), (short)0, (c), false, false)

// CDNA5 async global->LDS copy (ASYNCcnt-tracked), 16B per lane.
__device__ __forceinline__ void async_ld_b128(const float* g, uint32_t lds_byte_off) {
    asm volatile("global_load_async_to_lds_b128 %0, %1, off"
                 :: "v"(lds_byte_off), "v"(g) : "memory");
}
__device__ __forceinline__ void wait_async0() {
    asm volatile("s_wait_asynccnt 0x0" ::: "memory");
}
__device__ __forceinline__ uint32_t lds_off(const void* p) {
    return (uint32_t)(uintptr_t)p;   // low 32 bits of generic ptr = LDS byte address
}
// Runtime LDS-base select (no static pointer array -> no static addrspacecast).
__device__ __forceinline__ float* pick4(float* a, float* b, float* d, float* e, int ts) {
    float* p = a;
    if (ts == 1) p = b;
    if (ts == 2) p = d;
    if (ts == 3) p = e;
    return p;
}

// ---------------------------------------------------------------------------
// fp32 WMMA GEMM: C[M,N] = A[M,K] @ W[K,N] + bias[N]
// 128x128x32 tiles, 256 threads (8 waves, 32x64 each), async double-buffered LDS.
// Requires M%128==0, N%128==0, K%32==0.
// ---------------------------------------------------------------------------
__global__ __launch_bounds__(256) void gemm_bias_f32_wmma(
    const float* __restrict__ A, const float* __restrict__ W,
    const float* __restrict__ bias, float* __restrict__ C,
    int M, int N, int K)
{
    constexpr int BM = 128, BN = 128, BK = 32;
    constexpr int PA = 36;    // [m][k] pitch: even, 36*4=144 (16B-aligned rows)
    constexpr int PW = 132;   // [k][n] pitch: 132*4=528 (16B-aligned), 2*132%64=8 banks apart
    __shared__ float As[2][BM * PA];
    __shared__ float Ws[2][BK * PW];

    const int tid  = threadIdx.x;
    const int lane = tid & 31;
    const int wave = tid >> 5;
    const int m0 = blockIdx.y * BM;
    const int n0 = blockIdx.x * BN;
    const int wm = (wave & 3) * 32;
    const int wn = (wave >> 2) * 64;
    const int lm    = lane & 15;
    const int halfk = (lane >= 16) ? 2 : 0;
    const int rhalf = (lane >= 16) ? 8 : 0;

    auto stage = [&](int k0, int buf) {
        // A chunk 128x32: 1024 B128 transfers, 4 per thread.
        for (int i = 0; i < 4; ++i) {
            int idx = tid + i * 256;
            int r   = idx >> 3;
            int c4  = (idx & 7) * 4;
            async_ld_b128(A + (size_t)(m0 + r) * K + k0 + c4,
                          lds_off(&As[buf][r * PA + c4]));
        }
        // W chunk 32x128 (row-major [k][n]): 1024 B128 transfers, 4 per thread.
        for (int i = 0; i < 4; ++i) {
            int idx = tid + i * 256;
            int kk  = idx >> 5;
            int c4  = (idx & 31) * 4;
            async_ld_b128(W + (size_t)(k0 + kk) * N + n0 + c4,
                          lds_off(&Ws[buf][kk * PW + c4]));
        }
    };

    v8f acc[2][4];
    for (int mi = 0; mi < 2; ++mi)
        for (int ni = 0; ni < 4; ++ni)
            for (int i = 0; i < 8; ++i) acc[mi][ni][i] = 0.0f;

    stage(0, 0);
    int buf = 0;
    for (int k0 = 0; k0 < K; k0 += BK) {
        wait_async0();        // my async writes into As/Ws[buf] done
        __syncthreads();      // everyone's done; prev buffer free for reuse
        if (k0 + BK < K) stage(k0 + BK, buf ^ 1);   // prefetch next chunk

        const float* as = As[buf];
        const float* ws = Ws[buf];
        for (int kk = 0; kk < BK; kk += 4) {
            const int Kf = kk + halfk;
            v2f af[2], bf[4];
            for (int mi = 0; mi < 2; ++mi)
                af[mi] = *(const v2f*)&as[(wm + mi * 16 + lm) * PA + Kf];
            for (int ni = 0; ni < 4; ++ni) {
                const int col = wn + ni * 16 + lm;
                bf[ni].x = ws[Kf * PW + col];
                bf[ni].y = ws[(Kf + 1) * PW + col];
            }
            for (int mi = 0; mi < 2; ++mi)
                for (int ni = 0; ni < 4; ++ni)
                    acc[mi][ni] = WMMA_F32(af[mi], bf[ni], acc[mi][ni]);
        }
        buf ^= 1;
    }

    for (int mi = 0; mi < 2; ++mi)
        for (int ni = 0; ni < 4; ++ni) {
            const int col = n0 + wn + ni * 16 + lm;
            const float bcol = bias[col];
            for (int i = 0; i < 8; ++i) {
                const int row = m0 + wm + mi * 16 + i + rhalf;
                C[(size_t)row * N + col] = acc[mi][ni][i] + bcol;
            }
        }
}

// ---------------------------------------------------------------------------
// Fused local trittention: one block per (batch*head, window).
// scores[w,m,l] = sum_d c[w,d]*a_l[m,d]*b_l[l,d] via WMMA (A-frag = c ⊙ a_l)
// mask -> softmax -> marginals -> z = pm @ d_l + pl @ e_l
// ---------------------------------------------------------------------------
namespace cfg {
constexpr int T  = 2048, DM = 768, H = 12, DH = 64, W = 16, NW = 128, WW = 32;
constexpr int NCOL = 5 * DM;  // 3840
constexpr float IGNORE = -1.0e6f;
}

__global__ __launch_bounds__(256) void tritt_kernel(
    const float* __restrict__ abcde, float* __restrict__ z)
{
    using namespace cfg;
    constexpr int P = DH + 4;  // 68: even, 68*4=272 (16B-aligned rows), 68%64=4 banks
    __shared__ float c_s[W * P];
    __shared__ float a_s[WW * P], b_s[WW * P], d_s[WW * P], e_s[WW * P];
    __shared__ float pm_s[W][WW], pl_s[W][WW];

    const int bh = blockIdx.x / NW;
    const int n  = blockIdx.x % NW;
    const int bb = bh / H;
    const int hh = bh % H;
    const int tid  = threadIdx.x;
    const int lane = tid & 31;
    const int wave = tid >> 5;

    // Zero-fill look-back pad rows (n==0): rows 0..15, cols 0..63 of a/b/d/e.
    if (n == 0) {
        for (int i = 0; i < 4; ++i) {
            int idx = tid + i * 256;          // 0..1023
            int r   = idx >> 4;               // 0..63
            int c4  = (idx & 15) * 4;
            int ts  = r >> 4;                 // tensor 0..3
            int rr  = r & 15;                 // row 0..15
            float* s = pick4(a_s, b_s, d_s, e_s, ts) + rr * P + c4;
            s[0] = 0.0f; s[1] = 0.0f; s[2] = 0.0f; s[3] = 0.0f;
        }
    }

    // Async-stage c (16x64): 256 B128, one per thread.
    {
        int r  = tid >> 4;
        int c4 = (tid & 15) * 4;
        int t  = n * W + r;
        async_ld_b128(abcde + (size_t)(bb * T + t) * NCOL + 2 * DM + hh * DH + c4,
                      lds_off(&c_s[r * P + c4]));
    }
    // Async-stage look-around a,b,d,e (each 32x64): 512 B128 each, 2 per thread.
    for (int ts = 0; ts < 4; ++ts) {
        const int s_dm = (ts == 0) ? 0 : (ts == 1) ? 1 : (ts == 2) ? 3 : 4;
        float* dst = pick4(a_s, b_s, d_s, e_s, ts);
        for (int i = 0; i < 2; ++i) {
            int idx = tid + i * 256;
            int r   = idx >> 4;               // 0..31
            int c4  = (idx & 15) * 4;
            int t   = (n - 1) * W + r;
            if (t >= 0)                       // padded rows stay zero (disjoint LDS range)
                async_ld_b128(abcde + (size_t)(bb * T + t) * NCOL +
                                  s_dm * DM + hh * DH + c4,
                              lds_off(dst + r * P + c4));
        }
    }
    wait_async0();
    __syncthreads();

    const int lm    = lane & 15;
    const int halfk = (lane >= 16) ? 2 : 0;
    const int rhalf = (lane >= 16) ? 8 : 0;

    for (int wv = 0; wv < 2; ++wv) {
        const int w = wave * 2 + wv;     // query row, owned entirely by this wave
        const int q = n * W + w;

        // scores(32x32) = (c[w] ⊙ a_l)(32x64) @ b_l^T(64x32): 2x2 WMMA tiles.
        v8f sc[2][2];
        for (int mi = 0; mi < 2; ++mi)
            for (int ni = 0; ni < 2; ++ni)
                for (int i = 0; i < 8; ++i) sc[mi][ni][i] = 0.0f;

        for (int kk = 0; kk < DH; kk += 4) {
            const int Kf = kk + halfk;
            const v2f cf = *(const v2f*)&c_s[w * P + Kf];
            v2f af[2], bf[2];
            for (int mi = 0; mi < 2; ++mi)
                af[mi] = cf * (*(const v2f*)&a_s[(mi * 16 + lm) * P + Kf]);
            for (int ni = 0; ni < 2; ++ni)
                bf[ni] = *(const v2f*)&b_s[(ni * 16 + lm) * P + Kf];
            for (int mi = 0; mi < 2; ++mi)
                for (int ni = 0; ni < 2; ++ni)
                    sc[mi][ni] = WMMA_F32(af[mi], bf[ni], sc[mi][ni]);
        }

        // Causal look-around mask (matches reference incl. pad_value=0) + max.
        float smax = -3.0e38f;
        for (int mi = 0; mi < 2; ++mi)
            for (int ni = 0; ni < 2; ++ni)
                for (int i = 0; i < 8; ++i) {
                    const int m = mi * 16 + i + rhalf;
                    const int l = ni * 16 + lm;
                    const int pmv = (n == 0 && m < W) ? 0 : ((n - 1) * W + m);
                    const int plv = (n == 0 && l < W) ? 0 : ((n - 1) * W + l);
                    float s = sc[mi][ni][i];
                    if ((q < plv) || (plv <= pmv) || (s == 0.0f)) s = IGNORE;
                    sc[mi][ni][i] = s;
                    smax = fmaxf(smax, s);
                }
        for (int off = 16; off >= 1; off >>= 1)
            smax = fmaxf(smax, __shfl_xor(smax, off, 32));

        // softmax over the 1024 (m,l) entries of row w.
        float ssum = 0.0f;
        constexpr float invd = 1.0f / 64.0f;
        for (int mi = 0; mi < 2; ++mi)
            for (int ni = 0; ni < 2; ++ni)
                for (int i = 0; i < 8; ++i) {
                    float e = __expf((sc[mi][ni][i] - smax) * invd);
                    sc[mi][ni][i] = e;
                    ssum += e;
                }
        for (int off = 16; off >= 1; off >>= 1)
            ssum += __shfl_xor(ssum, off, 32);
        const float inv = 1.0f / ssum;

        // pl[l] = sum_m probs
        for (int ni = 0; ni < 2; ++ni) {
            float tsum = 0.0f;
            for (int mi = 0; mi < 2; ++mi)
                for (int i = 0; i < 8; ++i) tsum += sc[mi][ni][i];
            tsum += __shfl_xor(tsum, 16, 32);
            if (lane < 16) pl_s[w][ni * 16 + lm] = tsum * inv;
        }
        // pm[m] = sum_l probs
        for (int mi = 0; mi < 2; ++mi)
            for (int i = 0; i < 8; ++i) {
                float u = sc[mi][0][i] + sc[mi][1][i];
                u += __shfl_xor(u, 1, 32);
                u += __shfl_xor(u, 2, 32);
                u += __shfl_xor(u, 4, 32);
                u += __shfl_xor(u, 8, 32);
                if (lm == 0) pm_s[w][mi * 16 + i + rhalf] = u * inv;
            }

        // z[w,d] = sum_m pm[m]*d_l[m,d] + pl[m]*e_l[m,d] (wave-local LDS, in-order)
        float acc0 = 0.0f, acc1 = 0.0f;
        for (int m = 0; m < WW; ++m) {
            const float wm_ = pm_s[w][m];
            const float wl_ = pl_s[w][m];
            acc0 += wm_ * d_s[m * P + lane]      + wl_ * e_s[m * P + lane];
            acc1 += wm_ * d_s[m * P + lane + 32] + wl_ * e_s[m * P + lane + 32];
        }
        const int t = n * W + w;
        float* zp = z + (size_t)(bb * T + t) * (H * DH) + hh * DH;
        zp[lane]      = acc0;
        zp[lane + 32] = acc1;
    }
}

// ---------------------------------------------------------------------------
extern "C" void kernel_launch(void* const* d_in, const int* in_sizes, int n_in,
                              void* d_out, int out_size, void* d_ws, size_t ws_size,
                              hipStream_t stream) {
    (void)in_sizes; (void)n_in; (void)out_size; (void)ws_size;
    const float* x       = (const float*)d_in[0];
    const float* W_abcde = (const float*)d_in[1];
    const float* b_abcde = (const float*)d_in[2];
    const float* W_O     = (const float*)d_in[3];
    const float* b_O     = (const float*)d_in[4];
    float* out = (float*)d_out;

    constexpr int M = 2 * 2048;       // B*T = 4096
    constexpr int NCOL = 5 * 768;     // 3840
    constexpr int DMODEL = 768;

    float* abcde = (float*)d_ws;                 // 4096*3840 fp32
    float* zbuf  = abcde + (size_t)M * NCOL;     // 4096*768 fp32

    gemm_bias_f32_wmma<<<dim3(NCOL / 128, M / 128), 256, 0, stream>>>(
        x, W_abcde, b_abcde, abcde, M, NCOL, DMODEL);

    tritt_kernel<<<dim3(24 * 128), 256, 0, stream>>>(abcde, zbuf);

    gemm_bias_f32_wmma<<<dim3(DMODEL / 128, M / 128), 256, 0, stream>>>(
        zbuf, W_O, b_O, out, M, DMODEL, DMODEL);
}